// Sets2SetsLoss_73220602462532
// MI455X (gfx1250) — compile-verified
//
#include <hip/hip_runtime.h>

typedef __attribute__((ext_vector_type(2))) float v2f;
typedef __attribute__((ext_vector_type(4))) float v4f;
typedef __attribute__((ext_vector_type(8))) float v8f;

#define VOCAB      30000
#define BATCH      4096
#define SETSZ      64
#define LAMBDA_REG 10.0f
#define NTHREADS   256
#define NWAVES     (NTHREADS / 32)
#define CHUNK      128                      // elements per wave per main-loop iter
#define FULL       ((VOCAB / CHUNK) * CHUNK) // 29952: covered by the WMMA loop

// One workgroup (8 wave32) per batch row. Waves stream 128-element chunks of
// pred[row,:] with B128 non-temporal loads (491 MB single pass > 192 MB L2);
// each chunk is folded as two 16x4 f32 A-tiles into WMMA accumulators via
// D = A x ones(4x16) + C, so the reduction adds run on the matrix unit while
// the VALU/TRANS pipe does exp/square. The 48-element tail and the 64 targets
// (deduped in LDS) are handled with scalar code + ds_add_f32 atomics.
__global__ __launch_bounds__(NTHREADS)
void s2s_row_kernel(const float* __restrict__ pred,
                    const int*   __restrict__ target,
                    const float* __restrict__ weights,
                    float*       __restrict__ row_partial)
{
    const int row  = blockIdx.x;
    const int lane = threadIdx.x & 31;
    const int wave = threadIdx.x >> 5;
    const float* __restrict__ p = pred + (size_t)row * VOCAB;

    v8f c_sq = {};   // matrix-unit accumulator: sum of p^2
    v8f c_ex = {};   // matrix-unit accumulator: sum of exp(p)
    v2f ones; ones[0] = 1.0f; ones[1] = 1.0f;

    // Main loop: full chunks only -> no bounds checks, EXEC all-1s throughout.
    for (int base = wave * CHUNK; base + CHUNK <= VOCAB; base += CHUNK * NWAVES) {
        const v4f pv = __builtin_nontemporal_load(
            reinterpret_cast<const v4f*>(p + base + lane * 4));
        v2f sq0; sq0[0] = pv[0] * pv[0]; sq0[1] = pv[1] * pv[1];
        v2f sq1; sq1[0] = pv[2] * pv[2]; sq1[1] = pv[3] * pv[3];
        v2f ex0; ex0[0] = __expf(pv[0]); ex0[1] = __expf(pv[1]);
        v2f ex1; ex1[0] = __expf(pv[2]); ex1[1] = __expf(pv[3]);
        c_sq = __builtin_amdgcn_wmma_f32_16x16x4_f32(false, sq0, false, ones,
                                                     (short)0, c_sq, false, false);
        c_sq = __builtin_amdgcn_wmma_f32_16x16x4_f32(false, sq1, false, ones,
                                                     (short)0, c_sq, false, false);
        c_ex = __builtin_amdgcn_wmma_f32_16x16x4_f32(false, ex0, false, ones,
                                                     (short)0, c_ex, false, false);
        c_ex = __builtin_amdgcn_wmma_f32_16x16x4_f32(false, ex1, false, ones,
                                                     (short)0, c_ex, false, false);
    }

    // C layout: VGPR j holds row M=j (lanes 0-15) / M=j+8 (lanes 16-31); every
    // column identical (B was all ones). Per-lane sum of the 8 VGPRs gives
    // sum(rows 0-7) on the low half-wave, sum(rows 8-15) on the high half;
    // one xor-16 shuffle completes the wave total.
    float s_sq = c_sq[0] + c_sq[1] + c_sq[2] + c_sq[3]
               + c_sq[4] + c_sq[5] + c_sq[6] + c_sq[7];
    float s_ex = c_ex[0] + c_ex[1] + c_ex[2] + c_ex[3]
               + c_ex[4] + c_ex[5] + c_ex[6] + c_ex[7];
    s_sq += __shfl_xor(s_sq, 16, 32);
    s_ex += __shfl_xor(s_ex, 16, 32);

    // Scalar tail: elements [FULL, VOCAB) -- 48 of them, threads 0..47.
    float t_sq = 0.0f, t_ex = 0.0f;
    for (int v = FULL + (int)threadIdx.x; v < VOCAB; v += NTHREADS) {
        const float x = p[v];
        t_sq += x * x;
        t_ex += __expf(x);
    }

    __shared__ float sh_sq[NWAVES];
    __shared__ float sh_ex[NWAVES];
    __shared__ int   sh_t[SETSZ];
    __shared__ float sh_tsq;    // tail sum of p^2
    __shared__ float sh_tex;    // tail sum of exp(p)
    __shared__ float sh_pos;    // sum exp(-p[t]) over unique targets
    __shared__ float sh_negc;   // sum exp( p[t]) over unique targets
    __shared__ float sh_corr;   // sum (1 - 2*p[t]) over unique targets
    __shared__ int   sh_np;     // unique target count

    if (lane == 0) { sh_sq[wave] = s_sq; sh_ex[wave] = s_ex; }
    if (threadIdx.x == 0) {
        sh_tsq = 0.0f; sh_tex = 0.0f;
        sh_pos = 0.0f; sh_negc = 0.0f; sh_corr = 0.0f; sh_np = 0;
    }
    if (threadIdx.x < SETSZ) sh_t[threadIdx.x] = target[row * SETSZ + threadIdx.x];
    __syncthreads();

    if (FULL + (int)threadIdx.x < VOCAB) {
        atomicAdd(&sh_tsq, t_sq);
        atomicAdd(&sh_tex, t_ex);
    }

    // Dedup (duplicates in the scatter are idempotent): thread j owns target j,
    // contributes only if no earlier index holds the same vocab id.
    if (threadIdx.x < SETSZ) {
        const int t = sh_t[threadIdx.x];
        bool uniq = true;
        for (int k = 0; k < (int)threadIdx.x; ++k)
            if (sh_t[k] == t) { uniq = false; break; }
        if (uniq) {
            const float pt = p[t];
            atomicAdd(&sh_pos,  __expf(-pt));
            atomicAdd(&sh_negc, __expf(pt));
            atomicAdd(&sh_corr, 1.0f - 2.0f * pt);
            atomicAdd(&sh_np, 1);
        }
    }
    __syncthreads();

    if (threadIdx.x == 0) {
        float ssq = sh_tsq, sex = sh_tex;
        for (int w = 0; w < NWAVES; ++w) { ssq += sh_sq[w]; sex += sh_ex[w]; }
        const float wgt  = weights[row];
        const float npos = (float)sh_np;
        const float nneg = (float)VOCAB - npos;
        // MSE term: w * (sum p^2 + sum_unique (1 - 2 p[t]))
        const float mse  = wgt * (ssq + sh_corr);
        // Ranking term: pos_sum * (sum exp(p) - hot exp corr) / (np*nn)
        const float setl = sh_pos * (sex - sh_negc) / (npos * nneg);
        row_partial[row] = (mse + LAMBDA_REG * setl) * (1.0f / (float)BATCH);
    }
}

__global__ __launch_bounds__(256)
void s2s_reduce_kernel(const float* __restrict__ row_partial,
                       float* __restrict__ out)
{
    __shared__ float sh[256];
    float s = 0.0f;
    for (int i = threadIdx.x; i < BATCH; i += 256) s += row_partial[i];
    sh[threadIdx.x] = s;
    __syncthreads();
    for (int off = 128; off > 0; off >>= 1) {
        if (threadIdx.x < off) sh[threadIdx.x] += sh[threadIdx.x + off];
        __syncthreads();
    }
    if (threadIdx.x == 0) out[0] = sh[0];
}

extern "C" void kernel_launch(void* const* d_in, const int* in_sizes, int n_in,
                              void* d_out, int out_size, void* d_ws, size_t ws_size,
                              hipStream_t stream) {
    (void)in_sizes; (void)n_in; (void)out_size; (void)ws_size;
    const float* pred    = (const float*)d_in[0];
    const int*   target  = (const int*)d_in[1];
    const float* weights = (const float*)d_in[2];
    float*       out     = (float*)d_out;
    float*       rp      = (float*)d_ws;   // 4096 floats of scratch

    s2s_row_kernel<<<BATCH, NTHREADS, 0, stream>>>(pred, target, weights, rp);
    s2s_reduce_kernel<<<1, 256, 0, stream>>>(rp, out);
}